// EdgeModule_65146063946128
// MI455X (gfx1250) — compile-verified
//
#include <hip/hip_runtime.h>

typedef __attribute__((ext_vector_type(2))) float v2f;
typedef __attribute__((ext_vector_type(4))) float v4f;
typedef __attribute__((ext_vector_type(8))) float v8f;

#define Bv 2
#define Lv 1024
#define Dv 256
#define Ev 64
#define Tv 8

// ws layout in floats
#define OFF_ML 0
#define OFF_MR (Tv*Ev*Dv)                    // 131072
#define OFF_LP (2*Tv*Ev*Dv)                  // 262144
#define OFF_RP (2*Tv*Ev*Dv + Bv*Lv*Ev)      // 393216

// ---------------------------------------------------------------------------
// Kernel 1: M_left[t] = W1 @ L_t, M_right[t] = W2 @ R_t  via f32 WMMA.
//   M[t][e][d] = sum_m W[e][side*256+m] * Tr[t][m][d]
// grid = T * 2 sides * (E/16) * (D/16) = 1024 waves, blockDim = 32 (one wave).
// ---------------------------------------------------------------------------
__global__ void fuse_transform_kernel(const float* __restrict__ W_out,
                                      const float* __restrict__ left_tr,
                                      const float* __restrict__ right_tr,
                                      float* __restrict__ ws)
{
    const int lane  = threadIdx.x;            // 0..31
    const int bid   = blockIdx.x;             // 0..1023
    const int dTile = bid & 15;               // D/16 = 16 tiles
    const int eTile = (bid >> 4) & 3;         // E/16 = 4 tiles
    const int side  = (bid >> 6) & 1;         // 0=left, 1=right
    const int t     = bid >> 7;               // 0..7

    const float* Tr   = (side ? right_tr : left_tr) + t * Dv * Dv;
    const float* Wp   = W_out + side * Dv;    // row stride 2*Dv = 512
    float*       Mout = ws + (side ? OFF_MR : OFF_ML) + t * Ev * Dv;

    const int lm = lane & 15;                 // M (A) / N (B) index within tile
    const int kh = (lane >> 4) << 1;          // K sub-offset: lanes<16 -> {0,1}, lanes>=16 -> {2,3}

    const int eRow = eTile * 16 + lm;         // A-matrix row (M)
    const int dCol = dTile * 16 + lm;         // B-matrix col (N)

    v8f c = {0.f, 0.f, 0.f, 0.f, 0.f, 0.f, 0.f, 0.f};
    for (int k = 0; k < Dv; k += 4) {
        v2f a, b;
        // A: 16x4 slice of W  (M = eRow, K = k+kh, k+kh+1)
        a.x = Wp[eRow * (2 * Dv) + k + kh];
        a.y = Wp[eRow * (2 * Dv) + k + kh + 1];
        // B: 4x16 slice of Tr (K = k+kh, k+kh+1, N = dCol)
        b.x = Tr[(k + kh) * Dv + dCol];
        b.y = Tr[(k + kh + 1) * Dv + dCol];
        c = __builtin_amdgcn_wmma_f32_16x16x4_f32(
                /*neg_a=*/false, a, /*neg_b=*/false, b,
                /*c_mod=*/(short)0, c, /*reuse_a=*/false, /*reuse_b=*/false);
    }

    // C/D layout: VGPR v -> row v (lanes 0-15) or v+8 (lanes 16-31)
    const int eBase = eTile * 16 + ((lane >> 4) << 3);
#pragma unroll
    for (int v = 0; v < 8; ++v) {
        Mout[(eBase + v) * Dv + dCol] = c[v];
    }
}

// ---------------------------------------------------------------------------
// Kernel 2: per-token projection using the fused matrices (type gather).
//   lp[token][e] = M_left[t][e][:] . emb[token][:]  + b_out[e]
//   rp[token][e] = M_right[t][e][:] . emb[token][:]
// grid = B*L tokens, block = 128 (64 left lanes + 64 right lanes).
// ---------------------------------------------------------------------------
__global__ void token_proj_kernel(const float* __restrict__ embs,
                                  const int*   __restrict__ types,
                                  const float* __restrict__ b_out,
                                  float* __restrict__ ws)
{
    __shared__ v4f emb_s[Dv / 4];
    const int tid   = threadIdx.x;            // 0..127
    const int token = blockIdx.x;             // 0..2047

    if (tid < Dv / 4) {
        emb_s[tid] = ((const v4f*)(embs + token * Dv))[tid];
    }
    __syncthreads();

    const int t    = types[token];
    const int e    = tid & 63;
    const int side = tid >> 6;

    const v4f* M4 = (const v4f*)(ws + (side ? OFF_MR : OFF_ML) + (t * Ev + e) * Dv);

    float acc = 0.f;
#pragma unroll 4
    for (int d4 = 0; d4 < Dv / 4; ++d4) {
        v4f m = M4[d4];
        v4f x = emb_s[d4];
        acc += m.x * x.x + m.y * x.y + m.z * x.z + m.w * x.w;
    }

    if (side == 0) {
        ws[OFF_LP + token * Ev + e] = acc + b_out[e];
    } else {
        ws[OFF_RP + token * Ev + e] = acc;
    }
}

// ---------------------------------------------------------------------------
// Kernel 3: out[b][l][m][e] = lp[b][l][e] + rp[b][m][e]  (bias already in lp).
// Streaming 512MB write: float4 coalesced, non-temporal (bigger than L2).
// grid = B*L (one block per (b,l)), block = 256, 64 iters of 4KB stores.
// ---------------------------------------------------------------------------
__global__ void broadcast_add_kernel(const float* __restrict__ ws,
                                     float* __restrict__ out)
{
    __shared__ v4f lp_s[Ev / 4];
    const int tid = threadIdx.x;              // 0..255
    const int bid = blockIdx.x;               // b*L + l
    const int b   = bid >> 10;

    if (tid < Ev / 4) {
        lp_s[tid] = ((const v4f*)(ws + OFF_LP + bid * Ev))[tid];
    }
    __syncthreads();

    const v4f* rp4  = (const v4f*)(ws + OFF_RP) + (size_t)b * (Lv * Ev / 4);
    v4f*       out4 = (v4f*)out + (size_t)bid * (Lv * Ev / 4);

#pragma unroll 4
    for (int iter = 0; iter < (Lv * Ev / 4) / 256; ++iter) {
        const int idx4 = iter * 256 + tid;
        const int e4   = idx4 & 15;           // 16 float4 per m-row (E=64)
        const int m    = idx4 >> 4;
        v4f r = rp4[m * 16 + e4];             // L2-resident (512KB total)
        v4f l = lp_s[e4];
        v4f val = l + r;
        __builtin_nontemporal_store(val, &out4[idx4]);
    }
}

// ---------------------------------------------------------------------------
extern "C" void kernel_launch(void* const* d_in, const int* in_sizes, int n_in,
                              void* d_out, int out_size, void* d_ws, size_t ws_size,
                              hipStream_t stream) {
    const float* token_embs = (const float*)d_in[0];
    const int*   token_types = (const int*)d_in[1];
    const float* left_tr  = (const float*)d_in[2];
    const float* right_tr = (const float*)d_in[3];
    const float* W_out    = (const float*)d_in[4];
    const float* b_out    = (const float*)d_in[5];
    float* ws  = (float*)d_ws;
    float* out = (float*)d_out;

    fuse_transform_kernel<<<Tv * 2 * (Ev / 16) * (Dv / 16), 32, 0, stream>>>(
        W_out, left_tr, right_tr, ws);
    token_proj_kernel<<<Bv * Lv, 128, 0, stream>>>(
        token_embs, token_types, b_out, ws);
    broadcast_add_kernel<<<Bv * Lv, 256, 0, stream>>>(ws, out);
}